// SliceMamba_88175678587003
// MI455X (gfx1250) — compile-verified
//
#include <hip/hip_runtime.h>

// SliceMamba — exact dead-code-eliminated implementation for MI455X (gfx1250).
// Output depends ONLY on cls_token + weights (see analysis): one tiny fused
// latency-bound kernel, single workgroup of 16 wave32s, intermediates in LDS,
// GEMV tiles on V_WMMA_F32_16X16X4_F32 (full fp32 precision).

typedef __attribute__((ext_vector_type(2))) float v2f;
typedef __attribute__((ext_vector_type(8))) float v8f;

#define DMODEL 256
#define DIN    512
#define DTRANK 16
#define DSTATE 16
#define NCOUT  512
#define LN_EPS 1e-5f

__device__ __forceinline__ float silu_f(float x) {
    return x / (1.0f + expf(-x));
}
__device__ __forceinline__ float softplus_f(float x) {
    // jax.nn.softplus: log1p(exp(-|x|)) + max(x, 0)
    return fmaxf(x, 0.0f) + log1pf(expf(-fabsf(x)));
}

// 16 outputs: outlds[j0+n] = sum_k vec[k] * W[(j0+n)*K + k], K % 4 == 0.
// A(16x4) = vec chunk broadcast across all 16 M-rows; B(4x16) = W^T tile.
// ISA 32-bit A/B layout: lanes 0-15 hold K pair {0,1}, lanes 16-31 hold {2,3}.
// D row M=0 lives in accumulator VGPR0 on lanes 0-15.
__device__ __forceinline__ void gemv16_wmma(const float* __restrict__ W,
                                            const float* __restrict__ vec,
                                            int K, int j0,
                                            float* __restrict__ outlds) {
    const int lane = threadIdx.x & 31;
    const int half = lane >> 4;   // 0: K={0,1}, 1: K={2,3}
    const int col  = lane & 15;   // N (and M index for A broadcast rows)
    const float* wrow = W + (size_t)(j0 + col) * (size_t)K + 2 * half;
    v8f c = {0.f, 0.f, 0.f, 0.f, 0.f, 0.f, 0.f, 0.f};
    for (int k0 = 0; k0 < K; k0 += 4) {
        v2f a; a.x = vec[k0 + 2 * half];  a.y = vec[k0 + 2 * half + 1];
        v2f b; b.x = wrow[k0];            b.y = wrow[k0 + 1];
        c = __builtin_amdgcn_wmma_f32_16x16x4_f32(false, a, false, b,
                                                  (short)0, c, false, false);
    }
    if (half == 0) outlds[j0 + col] = c[0];
}

__global__ __launch_bounds__(512) void slicemamba_cls_kernel(
    const float* __restrict__ cls_token,   // [256]
    const float* __restrict__ in_proj_w,   // [1024, 256]
    const float* __restrict__ conv_w,      // [512, 1, 4]
    const float* __restrict__ conv_b,      // [512]
    const float* __restrict__ x_proj_w,    // [48, 512]
    const float* __restrict__ dt_proj_w,   // [512, 16]
    const float* __restrict__ dt_proj_b,   // [512]
    const float* __restrict__ D_skip,      // [512]
    const float* __restrict__ out_proj_w,  // [256, 512]
    const float* __restrict__ ln2_g,       // [256]
    const float* __restrict__ ln2_b,       // [256]
    const float* __restrict__ head_w,      // [512, 256]
    const float* __restrict__ head_b,      // [512]
    float* __restrict__ out)               // [512]
{
    __shared__ float s_cls[DMODEL];
    __shared__ float s_xz[2 * DIN];   // xc | zg
    __shared__ float s_xs[DIN];       // silu(conv(xc))
    __shared__ float s_zgs[DIN];      // silu(zg)
    __shared__ float s_xdbl[DTRANK + 2 * DSTATE];
    __shared__ float s_y[DIN];
    __shared__ float s_z[DMODEL];
    __shared__ float s_zn[DMODEL];
    __shared__ float s_red[2];

    const int tid  = threadIdx.x;
    const int wave = tid >> 5;        // 16 wave32s
    const int lane = tid & 31;

    if (tid < DMODEL) s_cls[tid] = cls_token[tid];
    __syncthreads();

    // ---- Stage A: xz[1024] = cls[256] @ in_proj_w^T  (WMMA, 4 tiles/wave)
    for (int t = 0; t < 4; ++t)
        gemv16_wmma(in_proj_w, s_cls, DMODEL, (wave + 16 * t) * 16, s_xz);
    __syncthreads();

    // ---- Stage B: causal conv @ t=0 (only last tap) + SiLU gates
    if (tid < DIN) {
        float xc    = s_xz[tid];
        float xconv = conv_w[tid * 4 + 3] * xc + conv_b[tid];
        s_xs[tid]   = silu_f(xconv);
        s_zgs[tid]  = silu_f(s_xz[DIN + tid]);
    }
    __syncthreads();

    // ---- Stage C: x_dbl[48] = xs[512] @ x_proj_w^T (wave-per-output reduce)
    for (int t = 0; t < 3; ++t) {
        int o = wave + 16 * t;                    // 0..47
        const float* r = x_proj_w + (size_t)o * DIN;
        float p = 0.0f;
        for (int i = lane; i < DIN; i += 32) p += s_xs[i] * r[i];
        for (int m = 16; m >= 1; m >>= 1) p += __shfl_xor(p, m, 32);
        if (lane == 0) s_xdbl[o] = p;
    }
    __syncthreads();

    // ---- Stage E: bc = Bss . Css (scalar; scan at t=0 collapses to this)
    if (tid == 0) {
        float bc = 0.0f;
        for (int s = 0; s < DSTATE; ++s)
            bc += s_xdbl[DTRANK + s] * s_xdbl[DTRANK + DSTATE + s];
        s_red[0] = bc;
    }
    __syncthreads();

    // ---- Stage D/F: delta = softplus(dt @ dt_proj_w^T + b); y = gated mix
    if (tid < DIN) {
        float acc = dt_proj_b[tid];
        const float* r = dt_proj_w + tid * DTRANK;
        for (int t = 0; t < DTRANK; ++t) acc += s_xdbl[t] * r[t];
        float delta = softplus_f(acc);
        float xs    = s_xs[tid];
        float ys    = delta * xs * s_red[0];      // h0=0 => ys0 = dBu0 . Css0
        s_y[tid]    = (ys + xs * D_skip[tid]) * s_zgs[tid];
    }
    __syncthreads();

    // ---- Stage G: z[256] = y[512] @ out_proj_w^T (WMMA, 1 tile/wave)
    gemv16_wmma(out_proj_w, s_y, DIN, wave * 16, s_z);
    __syncthreads();

    // ---- Stage H: LayerNorm over 256 (wave 0 reduces)
    if (wave == 0) {
        float sm = 0.0f, sq = 0.0f;
        for (int i = lane; i < DMODEL; i += 32) {
            float v = s_z[i]; sm += v; sq += v * v;
        }
        for (int m = 16; m >= 1; m >>= 1) {
            sm += __shfl_xor(sm, m, 32);
            sq += __shfl_xor(sq, m, 32);
        }
        if (lane == 0) {
            float mean = sm * (1.0f / DMODEL);
            float var  = sq * (1.0f / DMODEL) - mean * mean;
            s_red[0] = mean;
            s_red[1] = rsqrtf(var + LN_EPS);
        }
    }
    __syncthreads();
    if (tid < DMODEL)
        s_zn[tid] = (s_z[tid] - s_red[0]) * s_red[1] * ln2_g[tid] + ln2_b[tid];
    __syncthreads();

    // ---- Stage I: out[512] = zn[256] @ head_w^T + head_b (WMMA, 2 tiles/wave)
    for (int t = 0; t < 2; ++t) {
        int j0 = (wave + 16 * t) * 16;
        const int half = lane >> 4;
        const int col  = lane & 15;
        const float* wrow = head_w + (size_t)(j0 + col) * DMODEL + 2 * half;
        v8f c = {0.f, 0.f, 0.f, 0.f, 0.f, 0.f, 0.f, 0.f};
        for (int k0 = 0; k0 < DMODEL; k0 += 4) {
            v2f a; a.x = s_zn[k0 + 2 * half]; a.y = s_zn[k0 + 2 * half + 1];
            v2f b; b.x = wrow[k0];            b.y = wrow[k0 + 1];
            c = __builtin_amdgcn_wmma_f32_16x16x4_f32(false, a, false, b,
                                                      (short)0, c, false, false);
        }
        if (half == 0) out[j0 + col] = c[0] + head_b[j0 + col];
    }
}

extern "C" void kernel_launch(void* const* d_in, const int* in_sizes, int n_in,
                              void* d_out, int out_size, void* d_ws, size_t ws_size,
                              hipStream_t stream) {
    (void)in_sizes; (void)n_in; (void)d_ws; (void)ws_size; (void)out_size;
    // setup_inputs() order:
    //  0 dec4(dead) 1 patch_w(dead) 2 patch_b(dead) 3 row_emb(dead) 4 col_emb(dead)
    //  5 slice_emb(dead) 6 cls_token 7 ln1_g(dead) 8 ln1_b(dead) 9 in_proj_w
    // 10 conv_w 11 conv_b 12 x_proj_w 13 dt_proj_w 14 dt_proj_b 15 A_log(dead)
    // 16 D_skip 17 out_proj_w 18 ln2_g 19 ln2_b 20 head_w 21 head_b
    const float* cls_token  = (const float*)d_in[6];
    const float* in_proj_w  = (const float*)d_in[9];
    const float* conv_w     = (const float*)d_in[10];
    const float* conv_b     = (const float*)d_in[11];
    const float* x_proj_w   = (const float*)d_in[12];
    const float* dt_proj_w  = (const float*)d_in[13];
    const float* dt_proj_b  = (const float*)d_in[14];
    const float* D_skip     = (const float*)d_in[16];
    const float* out_proj_w = (const float*)d_in[17];
    const float* ln2_g      = (const float*)d_in[18];
    const float* ln2_b      = (const float*)d_in[19];
    const float* head_w     = (const float*)d_in[20];
    const float* head_b     = (const float*)d_in[21];
    float* out = (float*)d_out;

    slicemamba_cls_kernel<<<1, 512, 0, stream>>>(
        cls_token, in_proj_w, conv_w, conv_b, x_proj_w, dt_proj_w, dt_proj_b,
        D_skip, out_proj_w, ln2_g, ln2_b, head_w, head_b, out);
}